// TriangleCollisionLoss_84945863180903
// MI455X (gfx1250) — compile-verified
//
#include <hip/hip_runtime.h>
#include <hip/hip_bf16.h>

typedef __attribute__((ext_vector_type(2))) float v2f;
typedef __attribute__((ext_vector_type(8))) float v8f;

#define KNN 16
#define EPSV 0.001f

// ---------------- prep: pos, bary, sq, normals ----------------
__global__ void prep_kernel(const float* __restrict__ verts,
                            const int* __restrict__ faces,
                            float* __restrict__ pos,
                            float* __restrict__ bary,
                            float* __restrict__ sq,
                            float* __restrict__ nrm, int F) {
    int f = blockIdx.x * blockDim.x + threadIdx.x;
    if (f >= F) return;
    int i0 = faces[f * 3 + 0], i1 = faces[f * 3 + 1], i2 = faces[f * 3 + 2];
    float p0x = verts[i0 * 3 + 0], p0y = verts[i0 * 3 + 1], p0z = verts[i0 * 3 + 2];
    float p1x = verts[i1 * 3 + 0], p1y = verts[i1 * 3 + 1], p1z = verts[i1 * 3 + 2];
    float p2x = verts[i2 * 3 + 0], p2y = verts[i2 * 3 + 1], p2z = verts[i2 * 3 + 2];
    pos[f * 9 + 0] = p0x; pos[f * 9 + 1] = p0y; pos[f * 9 + 2] = p0z;
    pos[f * 9 + 3] = p1x; pos[f * 9 + 4] = p1y; pos[f * 9 + 5] = p1z;
    pos[f * 9 + 6] = p2x; pos[f * 9 + 7] = p2y; pos[f * 9 + 8] = p2z;
    float bx = (p0x + p1x + p2x) * (1.0f / 3.0f);
    float by = (p0y + p1y + p2y) * (1.0f / 3.0f);
    float bz = (p0z + p1z + p2z) * (1.0f / 3.0f);
    bary[f * 3 + 0] = bx; bary[f * 3 + 1] = by; bary[f * 3 + 2] = bz;
    sq[f] = bx * bx + by * by + bz * bz;
    float e1x = p1x - p0x, e1y = p1y - p0y, e1z = p1z - p0z;
    float e2x = p2x - p0x, e2y = p2y - p0y, e2z = p2z - p0z;
    float nx = e1y * e2z - e1z * e2y;
    float ny = e1z * e2x - e1x * e2z;
    float nz = e1x * e2y - e1y * e2x;
    float nl = sqrtf(nx * nx + ny * ny + nz * nz);
    nl = fmaxf(nl, 1e-12f);
    nrm[f * 3 + 0] = nx / nl; nrm[f * 3 + 1] = ny / nl; nrm[f * 3 + 2] = nz / nl;
}

__global__ void zero_kernel(float* __restrict__ fp, int F) {
    int i = blockIdx.x * blockDim.x + threadIdx.x;
    if (i < F) fp[i] = 0.0f;
}

// ---------------- knn: WMMA Gram tiles + threshold-filtered top-16 ----------------
// One block = 16 rows. 8 waves each produce one 16x16 d2 tile per 128-column
// chunk via V_WMMA_F32_16X16X4_F32 (K=3 padded to 4). Lanes filter their 8
// register-resident d2 values against the row's current 16th-best and push
// rare survivors to a per-row LDS queue (ds_add_rtn); 16 threads then do the
// sorted insert on survivors only. Survivor insertion is order-independent
// (a candidate >= the running 16th-best can never enter the list).
__global__ __launch_bounds__(256) void knn_kernel(const float* __restrict__ bary,
                                                  const float* __restrict__ sq,
                                                  int* __restrict__ nbr, int F) {
    __shared__ float qval[16][128];
    __shared__ int   qidx[16][128];
    __shared__ int   qcnt[16];
    __shared__ float thr[16];

    const int tid  = threadIdx.x;
    const int wave = tid >> 5;
    const int lane = tid & 31;
    const int lh   = lane & 15;
    const bool hi  = lane >= 16;
    const int row0 = blockIdx.x * 16;

    // A operand (rows): lanes 0-15 -> {x,y}; lanes 16-31 -> {z,0}
    int arow = row0 + lh; if (arow >= F) arow = F - 1;
    v2f a;
    {
        float ax = bary[arow * 3 + 0];
        float ay = bary[arow * 3 + 1];
        float az = bary[arow * 3 + 2];
        a.x = hi ? az : ax;
        a.y = hi ? 0.0f : ay;
    }

    // sq for the 8 rows this lane accumulates (C layout: +8 for high lanes)
    float sqr[8];
#pragma unroll
    for (int m = 0; m < 8; ++m) {
        int r = row0 + (hi ? 8 : 0) + m; if (r >= F) r = F - 1;
        sqr[m] = sq[r];
    }

    float bestv[KNN]; int besti[KNN];
#pragma unroll
    for (int k = 0; k < KNN; ++k) { bestv[k] = __builtin_inff(); besti[k] = 0; }

    if (tid < 16) { thr[tid] = __builtin_inff(); qcnt[tid] = 0; }
    __syncthreads();

    const int mbase = hi ? 8 : 0;
    const int lcol  = wave * 16 + lh;

    for (int cb = 0; cb < F; cb += 128) {
        int col  = cb + lcol;
        int colc = col < F ? col : F - 1;
        v2f b;
        {
            float bx = bary[colc * 3 + 0];
            float by = bary[colc * 3 + 1];
            float bz = bary[colc * 3 + 2];
            b.x = hi ? bz : bx;
            b.y = hi ? 0.0f : by;
        }
        float sqc = sq[colc];

        v8f c = {};
        c = __builtin_amdgcn_wmma_f32_16x16x4_f32(
                /*neg_a=*/false, a, /*neg_b=*/false, b,
                /*c_mod=*/(short)0, c, /*reuse_a=*/false, /*reuse_b=*/false);

        // Filter + push survivors (<=128 per row per chunk, so no overflow)
#pragma unroll
        for (int m = 0; m < 8; ++m) {
            float d2  = sqr[m] + sqc - 2.0f * c[m];
            int   row = mbase + m;
            if (col < F && d2 < thr[row]) {
                int slot = atomicAdd(&qcnt[row], 1);
                qval[row][slot] = d2;
                qidx[row][slot] = col;
            }
        }
        __syncthreads();

        if (tid < 16) {   // sorted insert on survivors only
            int n = qcnt[tid];
            for (int q = 0; q < n; ++q) {
                float d = qval[tid][q];
                if (d < bestv[KNN - 1]) {
                    int j = qidx[tid][q];
#pragma unroll
                    for (int p = KNN - 1; p >= 1; --p) {
                        float vp1 = bestv[p - 1];
                        if (d < vp1)           { bestv[p] = vp1; besti[p] = besti[p - 1]; }
                        else if (d < bestv[p]) { bestv[p] = d;   besti[p] = j; }
                    }
                    if (d < bestv[0]) { bestv[0] = d; besti[0] = j; }
                }
            }
            thr[tid]  = bestv[KNN - 1];
            qcnt[tid] = 0;
        }
        __syncthreads();
    }

    if (tid < 16 && (row0 + tid) < F) {
#pragma unroll
        for (int k = 0; k < KNN; ++k) nbr[(row0 + tid) * KNN + k] = besti[k];
    }
}

// ---------------- collide: plane/edge test + scatter-add ----------------
__global__ void collide_kernel(const float* __restrict__ pos,
                               const float* __restrict__ nrm,
                               const int* __restrict__ nbr,
                               float* __restrict__ fp, int F) {
    int idx = blockIdx.x * blockDim.x + threadIdx.x;
    if (idx >= F * KNN) return;
    int i = idx >> 4;
    int j = nbr[idx];
    float nx = nrm[i * 3 + 0], ny = nrm[i * 3 + 1], nz = nrm[i * 3 + 2];
    float px = pos[i * 9 + 0], py = pos[i * 9 + 1], pz = pos[i * 9 + 2];
    const float* tj = pos + (size_t)j * 9;
    bool anyhit = false;
#pragma unroll
    for (int e = 0; e < 3; ++e) {
        int en = (e == 2) ? 0 : e + 1;
        float l0x = tj[e * 3 + 0],  l0y = tj[e * 3 + 1],  l0z = tj[e * 3 + 2];
        float l1x = tj[en * 3 + 0], l1y = tj[en * 3 + 1], l1z = tj[en * 3 + 2];
        float denom = nx * (l1x - l0x) + ny * (l1y - l0y) + nz * (l1z - l0z);
        float numer = nx * (px - l0x) + ny * (py - l0y) + nz * (pz - l0z);
        float sd = (fabsf(denom) < EPSV) ? 1.0f : denom;
        float t  = numer / sd;
        anyhit = anyhit || ((fabsf(denom) >= EPSV) && (t > 0.0f) && (t < 1.0f));
    }
    if (anyhit && (j != i)) atomicAdd(&fp[j], 1.0f);   // exact integer-valued adds
}

// ---------------- reduce: deterministic weighted sum ----------------
__global__ __launch_bounds__(256) void reduce_kernel(const float* __restrict__ prob,
                                                     const float* __restrict__ fp,
                                                     float* __restrict__ out, int F) {
    __shared__ float s[256];
    int tid = threadIdx.x;
    float acc = 0.0f;
    for (int f = tid; f < F; f += 256) acc += prob[f] * fp[f];
    s[tid] = acc;
    __syncthreads();
    for (int off = 128; off > 0; off >>= 1) {
        if (tid < off) s[tid] += s[tid + off];
        __syncthreads();
    }
    if (tid == 0) out[0] = s[0] / (float)F;
}

extern "C" void kernel_launch(void* const* d_in, const int* in_sizes, int n_in,
                              void* d_out, int out_size, void* d_ws, size_t ws_size,
                              hipStream_t stream) {
    const float* verts = (const float*)d_in[0];
    const int*   faces = (const int*)d_in[1];
    const float* prob  = (const float*)d_in[2];
    const int F = in_sizes[1] / 3;

    float* ws   = (float*)d_ws;
    float* pos  = ws;                          // 9F
    float* bary = pos  + (size_t)9 * F;        // 3F
    float* sq   = bary + (size_t)3 * F;        // F
    float* nrm  = sq   + (size_t)F;            // 3F
    int*   nbr  = (int*)(nrm + (size_t)3 * F); // 16F ints
    float* fp   = (float*)(nbr + (size_t)KNN * F); // F

    int blocks = (F + 255) / 256;
    prep_kernel<<<blocks, 256, 0, stream>>>(verts, faces, pos, bary, sq, nrm, F);
    zero_kernel<<<blocks, 256, 0, stream>>>(fp, F);
    knn_kernel<<<(F + 15) / 16, 256, 0, stream>>>(bary, sq, nbr, F);
    collide_kernel<<<(F * KNN + 255) / 256, 256, 0, stream>>>(pos, nrm, nbr, fp, F);
    reduce_kernel<<<1, 256, 0, stream>>>(prob, fp, (float*)d_out, F);
}